// FastScatterW1_87153476370979
// MI455X (gfx1250) — compile-verified
//
#include <hip/hip_runtime.h>
#include <math.h>

typedef __attribute__((ext_vector_type(2))) float v2f;
typedef __attribute__((ext_vector_type(8))) float v8f;

#define NN   50000      // nodes
#define NE   800000     // edges
#define DD   64         // in channels
#define GG   100        // graphs
#define NPG  500        // nodes per graph
#define KORD 16
#define NCF  (KORD + 1) // 17 coeff rows
#define FEATC 704       // 11 * 64
#define OUTPG 2816      // 4 * 704

struct C4 { float v[4]; };
struct C6 { float v[6]; };

__device__ __forceinline__ void atomAddF(float* p, float v) {
  (void)__hip_atomic_fetch_add(p, v, __ATOMIC_RELAXED, __HIP_MEMORY_SCOPE_AGENT);
}

// ---------------- utility ----------------
__global__ void zero_f32(float* __restrict__ p, int n) {
  int i = blockIdx.x * blockDim.x + threadIdx.x;
  if (i < n) p[i] = 0.f;
}

__global__ void deg_kernel(const int* __restrict__ col, float* __restrict__ deg, int e) {
  int i = blockIdx.x * blockDim.x + threadIdx.x;
  if (i < e) atomAddF(&deg[col[i]], 1.0f);
}

__global__ void dh_kernel(float* __restrict__ deg, int n) {
  int i = blockIdx.x * blockDim.x + threadIdx.x;
  if (i < n) { float d = deg[i]; deg[i] = (d > 0.f) ? (1.0f / sqrtf(d)) : 0.f; }
}

__global__ void ew_kernel(const int* __restrict__ row, const int* __restrict__ col,
                          const float* __restrict__ dh, float* __restrict__ w, int e) {
  int i = blockIdx.x * blockDim.x + threadIdx.x;
  if (i < e) w[i] = -0.5f * dh[row[i]] * dh[col[i]];
}

// scatter: out[col[e], :] += w[e] * in[row[e], :]  (real edges only; self-loops folded out)
template <int FLOG2>
__global__ void scatter_kernel(const int* __restrict__ rows, const int* __restrict__ cols,
                               const float* __restrict__ w, const float* __restrict__ in,
                               float* __restrict__ out, int nEdges) {
  const int F = 1 << FLOG2;
  const int perEdge = F >> 2;                     // float4 chunks per edge
  long long tid = (long long)blockIdx.x * blockDim.x + threadIdx.x;
  long long total = (long long)nEdges * perEdge;
  if (tid >= total) return;
  int e = (int)(tid >> (FLOG2 - 2));
  int f = ((int)tid & (perEdge - 1)) << 2;
  float we = w[e];
  int r = rows[e], c = cols[e];
  const float4 hv = *(const float4*)(in + (long long)r * F + f);
  float* o = out + (long long)c * F + f;
  atomAddF(o + 0, we * hv.x);
  atomAddF(o + 1, we * hv.y);
  atomAddF(o + 2, we * hv.z);
  atomAddF(o + 3, we * hv.w);
}

// ---------------- cheb pass 1 (F = 64, 4 scale accumulators -> acc1 [N][4][64]) -------------
// All combine kernels re-zero tmp in place for the next scatter step.
__global__ void init1_kernel(const float* __restrict__ x, float* __restrict__ tmp,
                             float* __restrict__ t0, float* __restrict__ t1,
                             float* __restrict__ acc1, C4 c0, C4 c1) {
  int i = blockIdx.x * blockDim.x + threadIdx.x;
  if (i >= NN * DD) return;
  int n = i >> 6, d = i & 63;
  float xv = x[i];
  float t1v = 2.0f * tmp[i];       // T1 = 2*S(x)
  tmp[i] = 0.f;
  t0[i] = xv; t1[i] = t1v;
  float base = 0.5f * xv;          // coeffs[0] * 0.5 * T0
#pragma unroll
  for (int s = 0; s < 4; ++s)
    acc1[(long long)n * 256 + s * 64 + d] = c0.v[s] * base + c1.v[s] * t1v;
}

__global__ void step1_kernel(float* __restrict__ tmp, float* __restrict__ t0,
                             float* __restrict__ acc1, C4 ck) {
  int i = blockIdx.x * blockDim.x + threadIdx.x;
  if (i >= NN * DD) return;
  int n = i >> 6, d = i & 63;
  float t2 = 4.0f * tmp[i] - t0[i];   // T_k = 4*S(T_{k-1}) - T_{k-2}
  tmp[i] = 0.f;
  t0[i] = t2;                          // becomes new T_{k-1} after host pointer swap
#pragma unroll
  for (int s = 0; s < 4; ++s)
    acc1[(long long)n * 256 + s * 64 + d] += ck.v[s] * t2;
}

__global__ void abs_kernel(float* __restrict__ p, long long n) {
  long long i = (long long)blockIdx.x * blockDim.x + threadIdx.x;
  if (i < n) p[i] = fabsf(p[i]);
}

// ---------------- cheb pass 2 (F = 256, 6 FENG accumulators -> acc2 [N][6][64]) -------------
// FENG j: scale {1,2,2,3,3,3}, input block {0,0,1,0,1,2}
__global__ void init2_kernel(const float* __restrict__ h, float* __restrict__ tmp,
                             float* __restrict__ t0, float* __restrict__ t1,
                             float* __restrict__ acc2, C6 c0, C6 c1) {
  const int kBlk[6] = {0, 0, 1, 0, 1, 2};
  int i = blockIdx.x * blockDim.x + threadIdx.x;
  if (i >= NN * 256) return;
  int n = i >> 8, q = i & 255;
  int blk = q >> 6, d = q & 63;
  float hv = h[i];
  float t1v = 2.0f * tmp[i];
  tmp[i] = 0.f;
  t0[i] = hv; t1[i] = t1v;
  float base = 0.5f * hv;
#pragma unroll
  for (int j = 0; j < 6; ++j)
    if (kBlk[j] == blk)
      acc2[(long long)n * 384 + j * 64 + d] = c0.v[j] * base + c1.v[j] * t1v;
}

__global__ void step2_kernel(float* __restrict__ tmp, float* __restrict__ t0,
                             float* __restrict__ acc2, C6 ck) {
  const int kBlk[6] = {0, 0, 1, 0, 1, 2};
  int i = blockIdx.x * blockDim.x + threadIdx.x;
  if (i >= NN * 256) return;
  int n = i >> 8, q = i & 255;
  int blk = q >> 6, d = q & 63;
  float t2 = 4.0f * tmp[i] - t0[i];
  tmp[i] = 0.f;
  t0[i] = t2;
#pragma unroll
  for (int j = 0; j < 6; ++j)
    if (kBlk[j] == blk)
      acc2[(long long)n * 384 + j * 64 + d] += ck.v[j] * t2;
}

// ---------------- WMMA moments ----------------
// One wave (32 lanes) per (16-channel tile, 16-graph block).
// D = A(16x4 graph-selector) x B(4 nodes x 16 channels) + D accumulates segment sums.
// A is constant along K, so the result is the sum over the 4 nodes regardless of B's
// K permutation in the hardware layout. Source buffer selection is uniform per block
// (channel tiles never straddle feature sections); h and acc2 are already |.|.
__global__ __launch_bounds__(32)
void moments_kernel(const float* __restrict__ x, const float* __restrict__ h,
                    const float* __restrict__ a2, float* __restrict__ out) {
  int lane = threadIdx.x;
  int c0 = blockIdx.x * 16;          // 44 channel tiles
  int gbase = blockIdx.y * 16;       // 7 graph blocks
  int ngraphs = (GG - gbase) < 16 ? (GG - gbase) : 16;
  int numNodes = ngraphs * NPG;
  int nodeStart = gbase * NPG;
  int c = c0 + (lane & 15);
  int koff = (lane < 16) ? 0 : 2;

  // uniform (scalar) source selection
  const float* src; int stride, coff;
  if (c0 < 64)        { src = x;  stride = 64;  coff = c; }
  else if (c0 < 320)  { src = h;  stride = 256; coff = c - 64; }
  else                { src = a2; stride = 384; coff = c - 320; }
  const float* p = src + (long long)(nodeStart + koff) * stride + coff;
  const int step4 = 4 * stride;

  v8f S1 = {}, S2 = {}, S3 = {}, S4 = {};
  int g = 0, tcnt = 0;               // graph-in-block; 125 K-tiles per graph
  for (int k = 0; k < numNodes; k += 4) {
    float av = ((lane & 15) == g) ? 1.0f : 0.0f;
    v2f A = {av, av};
    float f0 = p[0];
    float f1 = p[stride];
    p += step4;
    v2f B  = {f0, f1};
    v2f B2 = {f0 * f0, f1 * f1};
    v2f B3 = {B2[0] * f0, B2[1] * f1};
    v2f B4 = {B2[0] * B2[0], B2[1] * B2[1]};
    S1 = __builtin_amdgcn_wmma_f32_16x16x4_f32(false, A, false, B,  (short)0, S1, false, false);
    S2 = __builtin_amdgcn_wmma_f32_16x16x4_f32(false, A, false, B2, (short)0, S2, false, false);
    S3 = __builtin_amdgcn_wmma_f32_16x16x4_f32(false, A, false, B3, (short)0, S3, false, false);
    S4 = __builtin_amdgcn_wmma_f32_16x16x4_f32(false, A, false, B4, (short)0, S4, false, false);
    if (++tcnt == NPG / 4) { tcnt = 0; ++g; }
  }

  const float invc = 1.0f / (float)NPG;
#pragma unroll
  for (int r = 0; r < 8; ++r) {
    int m = r + ((lane < 16) ? 0 : 8);
    int gg = gbase + m;
    if (gg >= GG) continue;
    float mean = S1[r] * invc, e2 = S2[r] * invc, e3 = S3[r] * invc, e4 = S4[r] * invc;
    float m2 = e2 - mean * mean;
    float m3 = e3 - 3.f * mean * e2 + 2.f * mean * mean * mean;
    float m4 = e4 - 4.f * mean * e3 + 6.f * mean * mean * e2 - 3.f * mean * mean * mean * mean;
    float skew = 0.f, kurt = -3.f;
    if (m2 > 0.f) {
      float sv = m3 / (m2 * sqrtf(m2)); skew = (sv > 1e15f) ? 0.f : sv;
      float kv = m4 / (m2 * m2) - 3.f;  kurt = (kv > 1e15f) ? -3.f : kv;
    }
    float* o = out + (long long)gg * OUTPG;
    o[c] = mean; o[704 + c] = m2; o[1408 + c] = skew; o[2112 + c] = kurt;
  }
}

// ---------------- host side ----------------
static void compute_coeffs(float cf[NCF][4]) {
  const int N = NCF;                 // 17
  const int scales[4] = {2, 4, 8, 16};
  double xs[NCF];
  for (int t = 0; t < N; ++t) xs[t] = cos(M_PI * (t + 0.5) / N) + 1.0;
  for (int si = 0; si < 4; ++si) {
    int s = scales[si];
    double ker[NCF];
    for (int t = 0; t < N; ++t) {
      double b = 1.0 - xs[t];
      double v = pow(b, (double)(s / 2)) - pow(b, (double)s);
      ker[t] = sqrt(v > 0.0 ? v : 0.0);
    }
    for (int o = 0; o < N; ++o) {
      double acc = 0.0;
      for (int t = 0; t < N; ++t) acc += ker[t] * cos(M_PI * o * (t + 0.5) / N);
      cf[o][si] = (float)(2.0 / N * acc);
    }
  }
}

static inline unsigned nb(long long n) { return (unsigned)((n + 255) / 256); }

extern "C" void kernel_launch(void* const* d_in, const int* in_sizes, int n_in,
                              void* d_out, int out_size, void* d_ws, size_t ws_size,
                              hipStream_t stream) {
  const float* x  = (const float*)d_in[0];
  const int*   ei = (const int*)d_in[1];
  const int* rows = ei;
  const int* cols = ei + NE;
  float* out = (float*)d_out;

  float* ws  = (float*)d_ws;
  float* deg  = ws;                        // 50048 (padded for alignment)
  float* w    = deg + 50048;               // 800000
  float* t0   = w + NE;                    // NN*256
  float* t1   = t0 + (long long)NN * 256;
  float* tmp  = t1 + (long long)NN * 256;
  float* hbuf = tmp + (long long)NN * 256; // acc1, then |.| in place (= s1 block of feat)
  float* acc2 = hbuf + (long long)NN * 256;// NN*384, |.| in place after pass 2

  float cf[NCF][4];
  compute_coeffs(cf);
  const int SCJ[6] = {1, 2, 2, 3, 3, 3};   // scale per FENG output

  dim3 b256(256), b32(32);

  // edge weights
  zero_f32<<<nb(NN), b256, 0, stream>>>(deg, NN);
  deg_kernel<<<nb(NE), b256, 0, stream>>>(cols, deg, NE);
  dh_kernel<<<nb(NN), b256, 0, stream>>>(deg, NN);
  ew_kernel<<<nb(NE), b256, 0, stream>>>(rows, cols, deg, w, NE);

  // ---- cheb pass 1 (F=64) ----
  zero_f32<<<nb((long long)NN * 64), b256, 0, stream>>>(tmp, NN * 64);
  scatter_kernel<6><<<nb((long long)NE * 16), b256, 0, stream>>>(rows, cols, w, x, tmp, NE);
  {
    C4 c0, c1;
    for (int s = 0; s < 4; ++s) { c0.v[s] = cf[0][s]; c1.v[s] = cf[1][s]; }
    init1_kernel<<<nb((long long)NN * 64), b256, 0, stream>>>(x, tmp, t0, t1, hbuf, c0, c1);
  }
  {
    float *p0 = t0, *p1 = t1;
    for (int k = 2; k <= KORD; ++k) {
      scatter_kernel<6><<<nb((long long)NE * 16), b256, 0, stream>>>(rows, cols, w, p1, tmp, NE);
      C4 ck; for (int s = 0; s < 4; ++s) ck.v[s] = cf[k][s];
      step1_kernel<<<nb((long long)NN * 64), b256, 0, stream>>>(tmp, p0, hbuf, ck);
      float* t = p0; p0 = p1; p1 = t;
    }
  }
  abs_kernel<<<nb((long long)NN * 256), b256, 0, stream>>>(hbuf, (long long)NN * 256);

  // ---- cheb pass 2 (F=256) ----
  zero_f32<<<nb((long long)NN * 256), b256, 0, stream>>>(tmp, NN * 256);
  scatter_kernel<8><<<nb((long long)NE * 64), b256, 0, stream>>>(rows, cols, w, hbuf, tmp, NE);
  {
    C6 c0, c1;
    for (int j = 0; j < 6; ++j) { c0.v[j] = cf[0][SCJ[j]]; c1.v[j] = cf[1][SCJ[j]]; }
    init2_kernel<<<nb((long long)NN * 256), b256, 0, stream>>>(hbuf, tmp, t0, t1, acc2, c0, c1);
  }
  {
    float *p0 = t0, *p1 = t1;
    for (int k = 2; k <= KORD; ++k) {
      scatter_kernel<8><<<nb((long long)NE * 64), b256, 0, stream>>>(rows, cols, w, p1, tmp, NE);
      C6 ck; for (int j = 0; j < 6; ++j) ck.v[j] = cf[k][SCJ[j]];
      step2_kernel<<<nb((long long)NN * 256), b256, 0, stream>>>(tmp, p0, acc2, ck);
      float* t = p0; p0 = p1; p1 = t;
    }
  }
  abs_kernel<<<nb((long long)NN * 384), b256, 0, stream>>>(acc2, (long long)NN * 384);

  // ---- per-graph moments via WMMA f32 16x16x4 ----
  dim3 mgrid(FEATC / 16, (GG + 15) / 16);  // 44 x 7
  moments_kernel<<<mgrid, b32, 0, stream>>>(x, hbuf, acc2, out);
}